// SEWGCN_10402410791110
// MI455X (gfx1250) — compile-verified
//
#include <hip/hip_runtime.h>

#define NN 100000
#define EE 1000000
#define NF 128
#define NH 256
#define NC 40
#define NCP 48   // NC padded to a multiple of 16

typedef __attribute__((ext_vector_type(2))) float v2f;
typedef __attribute__((ext_vector_type(8))) float v8f;

__device__ __forceinline__ float wave_reduce_sum(float v) {
  // wave32 reduction (CDNA5 is wave32-only)
  for (int off = 16; off > 0; off >>= 1)
    v += __shfl_down(v, off, 32);
  return v;
}

// ---- per-node L2 norm: one wave per node -------------------------------
__global__ void sew_node_norm(const float* __restrict__ feat, float* __restrict__ nrm,
                              int n, int d) {
  int wid = threadIdx.x >> 5, lane = threadIdx.x & 31;
  int i = blockIdx.x * (blockDim.x >> 5) + wid;
  if (i >= n) return;
  const float* f = feat + (size_t)i * d;
  float acc = 0.f;
  for (int j = lane; j < d; j += 32) { float v = f[j]; acc += v * v; }
  acc = wave_reduce_sum(acc);
  if (lane == 0) nrm[i] = fmaxf(sqrtf(acc), 1e-8f);
}

// ---- per-edge cosine similarity + threshold mask: one wave per edge ----
__global__ void sew_edge_att(const float* __restrict__ feat, const int* __restrict__ row,
                             const int* __restrict__ col, const float* __restrict__ nrm,
                             const float* __restrict__ prevw, float* __restrict__ w,
                             int e_count, int d) {
  int wid = threadIdx.x >> 5, lane = threadIdx.x & 31;
  int e = blockIdx.x * (blockDim.x >> 5) + wid;
  if (e >= e_count) return;
  int r = row[e], c = col[e];
  const float* fr = feat + (size_t)r * d;
  const float* fc = feat + (size_t)c * d;
  float acc = 0.f;
  for (int j = lane; j < d; j += 32) acc += fr[j] * fc[j];
  acc = wave_reduce_sum(acc);
  if (lane == 0) {
    float sim = acc / (nrm[r] * nrm[c]);
    bool m = (sim >= 0.1f) && (r != c);
    if (prevw) m = m && (prevw[e] > 0.f);   // mask1 <=> w1 > 0 (sims >= 0.1 > 0)
    w[e] = m ? sim : 0.f;
  }
}

// ---- weighted degree + rsqrt(deg+1) ------------------------------------
__global__ void sew_deg_accum(const int* __restrict__ row, const float* __restrict__ w,
                              float* __restrict__ deg, int e_count) {
  int e = blockIdx.x * blockDim.x + threadIdx.x;
  if (e >= e_count) return;
  float we = w[e];
  if (we != 0.f) atomicAdd(&deg[row[e]], we);
}

__global__ void sew_dinv(float* __restrict__ deg, int n) {
  int i = blockIdx.x * blockDim.x + threadIdx.x;
  if (i >= n) return;
  deg[i] = rsqrtf(deg[i] + 1.0f);   // + self-loop weight 1
}

// ---- zero-pad W2 [NH x NC] -> Bp [NH x NCP] ----------------------------
__global__ void sew_pad_w2(const float* __restrict__ W2, float* __restrict__ Bp) {
  int idx = blockIdx.x * blockDim.x + threadIdx.x;
  if (idx >= NH * NCP) return;
  int k = idx / NCP, c = idx % NCP;
  Bp[idx] = (c < NC) ? W2[k * NC + c] : 0.f;
}

// ---- f32 WMMA GEMM: C[M,N] = A[M,K] @ B[K,N], exact tiles, no guards ---
// Block = 256 threads (8 waves). The block owns a TN-wide column strip;
// the K x TN B-tile is staged in LDS once (row-PAIRED layout: element p*TN+c
// holds {B[2p][c], B[2p+1][c]}) and shared by 8 M-tiles (one per wave), so
// each lane's B fragment is a single aligned ds_load_b64 landing directly in
// an even-aligned VGPR pair. WMMA runs under a per-wave-uniform branch =>
// EXEC all-1s as the ISA requires.
// A fragment (16x4 f32): lanes 0-15 M=lane K={0,1}; lanes 16-31 K={2,3}
// B fragment (4x16 f32): lane&15 = N col; half-wave selects K row pair
// C (16x16 f32, 8 VGPRs): VGPR v -> M = v + 8*(lane>=16); N = lane&15
template <int TN>
__launch_bounds__(256)
__global__ void sew_gemm_wmma(const float* __restrict__ A, const float* __restrict__ B,
                              float* __restrict__ C, int mTiles, int N, int K) {
  constexpr int NT = TN / 16;
  extern __shared__ float smem[];              // K * TN floats = (K/2)*TN float2
  v2f* __restrict__ smv = (v2f*)smem;
  const int tid = threadIdx.x;
  const int wid = tid >> 5, lane = tid & 31;
  const int m = lane & 15;
  const int half = lane >> 4;                  // 0 or 1 -> K rows {0,1} / {2,3}
  const int colBase = blockIdx.y * TN;

  // cooperative stage of the B tile in row-paired layout
  for (int i = tid; i < (K / 2) * TN; i += 256) {
    const int p = i / TN, cc = i % TN;
    v2f pr;
    pr.x = B[(size_t)(2 * p) * N + colBase + cc];
    pr.y = B[(size_t)(2 * p + 1) * N + colBase + cc];
    smv[i] = pr;                               // ds_store_b64
  }
  __syncthreads();                             // s_barrier_signal/wait on gfx1250

  const int tileM = blockIdx.x * 8 + wid;
  if (tileM < mTiles) {                        // uniform across the wave
    v8f acc[NT] = {};
    const float* arow = A + (size_t)((size_t)tileM * 16 + m) * K;
    __builtin_prefetch(arow + 64, 0, 0);       // global_prefetch_b8
    const int bIdx0 = half * TN + m;           // lane-constant part of B index
    for (int k0 = 0; k0 < K; k0 += 4) {
      const int kr = k0 + half * 2;
      const float2 av = *(const float2*)(arow + kr);
      v2f a; a.x = av.x; a.y = av.y;
      const int bRow = (k0 >> 1) * TN + bIdx0;
#pragma unroll
      for (int t = 0; t < NT; ++t) {
        const v2f b = smv[bRow + t * 16];      // ds_load_b64, even-aligned pair
        acc[t] = __builtin_amdgcn_wmma_f32_16x16x4_f32(
            /*neg_a=*/false, a, /*neg_b=*/false, b,
            /*c_mod=*/(short)0, acc[t], /*reuse_a=*/false, /*reuse_b=*/false);
      }
    }
    const size_t rowBase = (size_t)tileM * 16 + (size_t)half * 8;
#pragma unroll
    for (int t = 0; t < NT; ++t) {
      const int cc = colBase + t * 16 + m;
#pragma unroll
      for (int v = 0; v < 8; ++v)
        C[(rowBase + v) * N + cc] = acc[t][v];
    }
  }
}

// ---- edge aggregation: out[row] += dinv[row]*w*dinv[col] * h[col] ------
__global__ void sew_agg(const int* __restrict__ row, const int* __restrict__ col,
                        const float* __restrict__ w, const float* __restrict__ dinv,
                        const float* __restrict__ h, float* __restrict__ out,
                        int e_count, int d, int ldH, int ldO) {
  int wid = threadIdx.x >> 5, lane = threadIdx.x & 31;
  int e = blockIdx.x * (blockDim.x >> 5) + wid;
  if (e >= e_count) return;
  float we = w[e];
  if (we == 0.f) return;                 // zero-weight edge == removed edge
  int r = row[e], c = col[e];
  float nv = dinv[r] * we * dinv[c];
  const float* hc = h + (size_t)c * ldH;
  float* orow = out + (size_t)r * ldO;
  for (int f = lane; f < d; f += 32)
    atomicAdd(&orow[f], nv * hc[f]);
}

// ---- h = relu(agg + dinv^2 * h0 + b)  (layer 1, in-place into h0) ------
__global__ void sew_combine_relu(float* __restrict__ h0, const float* __restrict__ agg,
                                 const float* __restrict__ dinv, const float* __restrict__ bias,
                                 int n, int d) {
  size_t idx = (size_t)blockIdx.x * blockDim.x + threadIdx.x;
  if (idx >= (size_t)n * d) return;
  int i = (int)(idx / d), f = (int)(idx % d);
  float di = dinv[i];
  float v = agg[idx] + di * di * h0[idx] + bias[f];
  h0[idx] = fmaxf(v, 0.f);
}

// ---- out += dinv^2 * h2 + b  (layer 2; h2 has leading dim NCP) ---------
__global__ void sew_combine_out(float* __restrict__ out, const float* __restrict__ h2,
                                const float* __restrict__ dinv, const float* __restrict__ bias,
                                int n) {
  size_t idx = (size_t)blockIdx.x * blockDim.x + threadIdx.x;
  if (idx >= (size_t)n * NC) return;
  int i = (int)(idx / NC), f = (int)(idx % NC);
  float di = dinv[i];
  out[idx] = out[idx] + di * di * h2[(size_t)i * NCP + f] + bias[f];
}

extern "C" void kernel_launch(void* const* d_in, const int* in_sizes, int n_in,
                              void* d_out, int out_size, void* d_ws, size_t ws_size,
                              hipStream_t stream) {
  (void)in_sizes; (void)n_in; (void)out_size; (void)ws_size;
  const float* x  = (const float*)d_in[0];
  const int*   row = (const int*)d_in[1];
  const int*   col = (const int*)d_in[2];
  const float* W1 = (const float*)d_in[3];
  const float* b1 = (const float*)d_in[4];
  const float* W2 = (const float*)d_in[5];
  const float* b2 = (const float*)d_in[6];
  float* out = (float*)d_out;

  char* ws = (char*)d_ws;
  size_t off = 0;
  float* h0    = (float*)(ws + off); off += (size_t)NN * NH * sizeof(float);
  float* agg1  = (float*)(ws + off); off += (size_t)NN * NH * sizeof(float);
  float* w1    = (float*)(ws + off); off += (size_t)EE * sizeof(float);
  float* w2    = (float*)(ws + off); off += (size_t)EE * sizeof(float);
  float* nrm   = (float*)(ws + off); off += (size_t)NN * sizeof(float);
  float* dinv1 = (float*)(ws + off); off += (size_t)NN * sizeof(float);
  float* dinv2 = (float*)(ws + off); off += (size_t)NN * sizeof(float);
  float* W2p   = (float*)(ws + off); off += (size_t)NH * NCP * sizeof(float);
  float* h2 = agg1;  // agg1 is dead after combine_relu; reuse for padded h@W2

  // zero accumulators every call (graph-capture-safe async memsets)
  hipMemsetAsync(agg1,  0, (size_t)NN * NH * sizeof(float), stream);
  hipMemsetAsync(dinv1, 0, (size_t)NN * sizeof(float), stream);
  hipMemsetAsync(dinv2, 0, (size_t)NN * sizeof(float), stream);
  hipMemsetAsync(out,   0, (size_t)NN * NC * sizeof(float), stream);

  const int WPB = 8;            // wave32 waves per 256-thread block
  const int mTiles = NN / 16;   // 6250 exact 16-row tiles
  const int gemmBlocks = (mTiles + 7) / 8;   // 782

  sew_pad_w2<<<(NH * NCP + 255) / 256, 256, 0, stream>>>(W2, W2p);

  // ---- layer 1 ----
  sew_node_norm<<<(NN + WPB - 1) / WPB, 256, 0, stream>>>(x, nrm, NN, NF);
  sew_edge_att<<<(EE + WPB - 1) / WPB, 256, 0, stream>>>(x, row, col, nrm, nullptr, w1, EE, NF);
  sew_deg_accum<<<(EE + 255) / 256, 256, 0, stream>>>(row, w1, dinv1, EE);
  sew_dinv<<<(NN + 255) / 256, 256, 0, stream>>>(dinv1, NN);
  sew_gemm_wmma<64><<<dim3(gemmBlocks, NH / 64), 256, (size_t)NF * 64 * sizeof(float), stream>>>(
      x, W1, h0, mTiles, NH, NF);
  sew_agg<<<(EE + WPB - 1) / WPB, 256, 0, stream>>>(row, col, w1, dinv1, h0, agg1, EE, NH, NH, NH);
  sew_combine_relu<<<(int)(((size_t)NN * NH + 255) / 256), 256, 0, stream>>>(h0, agg1, dinv1, b1, NN, NH);

  // ---- layer 2 ----
  sew_node_norm<<<(NN + WPB - 1) / WPB, 256, 0, stream>>>(h0, nrm, NN, NH);
  sew_edge_att<<<(EE + WPB - 1) / WPB, 256, 0, stream>>>(h0, row, col, nrm, w1, w2, EE, NH);
  sew_deg_accum<<<(EE + 255) / 256, 256, 0, stream>>>(row, w2, dinv2, EE);
  sew_dinv<<<(NN + 255) / 256, 256, 0, stream>>>(dinv2, NN);
  sew_gemm_wmma<NCP><<<dim3(gemmBlocks, 1), 256, (size_t)NH * NCP * sizeof(float), stream>>>(
      h0, W2p, h2, mTiles, NCP, NH);
  sew_agg<<<(EE + WPB - 1) / WPB, 256, 0, stream>>>(row, col, w2, dinv2, h2, out, EE, NC, NCP, NC);
  sew_combine_out<<<(int)(((size_t)NN * NC + 255) / 256), 256, 0, stream>>>(out, h2, dinv2, b2, NN);
}